// PenaltySampleHead_2774548873613
// MI455X (gfx1250) — compile-verified
//
#include <hip/hip_runtime.h>
#include <cstdint>

#define NB       128          // batch
#define NV       128000       // vocab
#define NH       200          // history length
#define TOPK     50
#define MINKEEP  5
#define NBUCK    4096
#define NTHREADS 256
#define NWAVES   (NTHREADS / 32)

#define SPLIT    20
#define SEG      (NV / SPLIT)       // 6400 floats = 25.6 KB per segment
#define NQ       (SEG / 4)          // 1600 quads
#define NCHUNK   (SEG / 128)        // 50 async B128 wave-chunks
#define MAXCS    192                // per-segment candidate cap (expect ~90)
#define NSEGBLK  (NB * SPLIT)       // 2560 workgroups in pass kernel
#define MERGEN   (SPLIT * TOPK)     // 1000 candidates per row at merge

typedef unsigned long long u64;

// monotone order-preserving map: float -> unsigned (ascending)
__device__ __forceinline__ unsigned fkey(float f) {
    unsigned u = __float_as_uint(f);
    return (u & 0x80000000u) ? ~u : (u | 0x80000000u);
}
__device__ __forceinline__ float fkey_inv(unsigned k) {
    unsigned u = (k & 0x80000000u) ? (k ^ 0x80000000u) : ~k;
    return __uint_as_float(u);
}
// packed candidate: value-order in hi 32, jax tie-break (lower vocab idx wins) in lo 32
__device__ __forceinline__ u64 pack_cand(float v, int gidx) {
    return ((u64)fkey(v) << 32) | (unsigned)(NV - 1 - gidx);
}

// ---------------- kernel A: per-segment exact top-50 ----------------
__global__ __launch_bounds__(NTHREADS)
void seg_topk_kernel(const float* __restrict__ logits,
                     const int*   __restrict__ ids,
                     const float* __restrict__ penalty,
                     u64* __restrict__ pws)
{
    __shared__ __align__(16) float s_stage[SEG];     // 25600 B, whole segment staged
    __shared__ unsigned s_hist[NBUCK];               // 16384 B
    __shared__ unsigned s_bitmap[SEG / 32];          // 800 B
    __shared__ unsigned s_bsum[NTHREADS];            // 1024 B
    __shared__ u64      s_cpk[MAXCS];                // 1536 B
    __shared__ int      s_nc;
    __shared__ int      s_bstar;

    const int row     = blockIdx.x / SPLIT;
    const int seg     = blockIdx.x % SPLIT;
    const int segBase = seg * SEG;
    const int tid  = threadIdx.x;
    const int lane = tid & 31;
    const int wave = tid >> 5;
    const float pen = penalty[row];
    const float* segp = logits + (size_t)row * NV + segBase;

    // ---- kick off DMA of the whole segment into LDS first (deep async pipeline) ----
    for (int c = wave; c < NCHUNK; c += NWAVES) {
        const float* g = segp + c * 128 + lane * 4;
        unsigned lds = (unsigned)(uintptr_t)(&s_stage[c * 128 + lane * 4]);
        asm volatile("global_load_async_to_lds_b128 %0, %1, off"
                     :: "v"(lds), "v"(g) : "memory");
    }

    // ---- overlap: zero histogram/bitmap, build penalty bitmap for this segment ----
    for (int i = tid; i < SEG / 32; i += NTHREADS) s_bitmap[i] = 0u;
    for (int i = tid; i < NBUCK;   i += NTHREADS) s_hist[i]   = 0u;
    if (tid == 0) s_nc = 0;
    __syncthreads();

    for (int t = tid; t < NH; t += NTHREADS) {
        int id = ids[row * NH + t] - segBase;
        if ((unsigned)id < (unsigned)SEG)
            atomicOr(&s_bitmap[id >> 5], 1u << (id & 31));
    }
    asm volatile("s_wait_asynccnt 0x0" ::: "memory");
    __syncthreads();   // staged data + bitmap visible to all waves

    // ---- 12-bit-key histogram of penalty-modified values (from LDS) ----
    for (int q = tid; q < NQ; q += NTHREADS) {
        float4 v4 = *(const float4*)&s_stage[q * 4];
        unsigned wb = s_bitmap[q >> 3];              // quad lies within one word
        float vv[4] = { v4.x, v4.y, v4.z, v4.w };
        #pragma unroll
        for (int c = 0; c < 4; ++c) {
            float v = vv[c];
            if (wb & (1u << ((q * 4 + c) & 31)))
                v = (v < 0.0f) ? v * pen : v / pen;
            atomicAdd(&s_hist[fkey(v) >> 20], 1u);
        }
    }
    __syncthreads();

    // ---- threshold bucket b*: largest b with count(key >= b) >= TOPK ----
    {
        const int blk = NBUCK / NTHREADS;            // 16 buckets/thread, from the top
        const int hiB = NBUCK - blk * tid;
        unsigned sum = 0;
        for (int j = hiB - blk; j < hiB; ++j) sum += s_hist[j];
        s_bsum[tid] = sum;
        __syncthreads();
        for (int off = 1; off < NTHREADS; off <<= 1) {
            unsigned x = (tid >= off) ? s_bsum[tid - off] : 0u;
            __syncthreads();
            s_bsum[tid] += x;
            __syncthreads();
        }
        unsigned inc = s_bsum[tid];
        unsigned exc = tid ? s_bsum[tid - 1] : 0u;
        if (inc >= TOPK && exc < TOPK) {
            unsigned cum = exc;
            for (int j = hiB - 1; j >= hiB - blk; --j) {
                cum += s_hist[j];
                if (cum >= TOPK) { s_bstar = j; break; }
            }
        }
        __syncthreads();
    }
    const unsigned bstar = (unsigned)s_bstar;

    // ---- collect candidates (key >= b*) from staged LDS, pre-packed as u64 ----
    for (int q = tid; q < NQ; q += NTHREADS) {
        float4 v4 = *(const float4*)&s_stage[q * 4];
        unsigned wb = s_bitmap[q >> 3];
        float vv[4] = { v4.x, v4.y, v4.z, v4.w };
        #pragma unroll
        for (int c = 0; c < 4; ++c) {
            float v = vv[c];
            if (wb & (1u << ((q * 4 + c) & 31)))
                v = (v < 0.0f) ? v * pen : v / pen;
            if ((fkey(v) >> 20) >= bstar) {
                int slot = atomicAdd(&s_nc, 1);
                if (slot < MAXCS) s_cpk[slot] = pack_cand(v, segBase + q * 4 + c);
            }
        }
    }
    __syncthreads();

    // ---- wave 0: exact sorted top-50 of segment -> global scratch (packed) ----
    const int nc = min(s_nc, MAXCS);
    if (wave == 0) {
        u64* outp = pws + (size_t)blockIdx.x * TOPK;
        for (int k = 0; k < TOPK; ++k) {
            u64 best = 0;
            for (int j = lane; j < nc; j += 32) { u64 p = s_cpk[j]; best = p > best ? p : best; }
            #pragma unroll
            for (int off = 16; off > 0; off >>= 1) {
                u64 o = __shfl_xor(best, off, 32);
                best = o > best ? o : best;
            }
            if (lane == 0) outp[k] = best;
            for (int j = lane; j < nc; j += 32)   // kill winner (packed keys are unique)
                if (s_cpk[j] == best) s_cpk[j] = 0;
            __builtin_amdgcn_wave_barrier();
        }
    }
}

// ---------------- kernel B: merge 20x50 per row + softmax/top-p tail ----------------
__global__ __launch_bounds__(NTHREADS)
void merge_finalize_kernel(const u64* __restrict__ pws,
                           const float* __restrict__ top_p,
                           const float* __restrict__ temperature,
                           float* __restrict__ out)
{
    __shared__ u64   s_pk[MERGEN];           // 1000 packed candidates
    __shared__ u64   s_pk2[NWAVES * TOPK];   // 400 stage-1 winners
    __shared__ float s_selv[TOPK];
    __shared__ int   s_selt[TOPK];
    __shared__ float s_e[TOPK];
    __shared__ int   s_mk[TOPK];

    const int row  = blockIdx.x;
    const int tid  = threadIdx.x;
    const int lane = tid & 31;
    const int wave = tid >> 5;

    for (int i = tid; i < MERGEN; i += NTHREADS)
        s_pk[i] = pws[(size_t)row * MERGEN + i];
    __syncthreads();

    // stage 1: each wave -> top-50 of its 125-entry slice
    {
        const int lo = wave * (MERGEN / NWAVES);   // 125 per wave, exact
        const int hi = lo + (MERGEN / NWAVES);
        for (int k = 0; k < TOPK; ++k) {
            u64 best = 0;
            for (int j = lo + lane; j < hi; j += 32) { u64 p = s_pk[j]; best = p > best ? p : best; }
            #pragma unroll
            for (int off = 16; off > 0; off >>= 1) {
                u64 o = __shfl_xor(best, off, 32);
                best = o > best ? o : best;
            }
            if (lane == 0) s_pk2[wave * TOPK + k] = best;
            for (int j = lo + lane; j < hi; j += 32)
                if (s_pk[j] == best) s_pk[j] = 0;
            __builtin_amdgcn_wave_barrier();
        }
    }
    __syncthreads();

    // stage 2: wave 0 -> global top-50 of the 400 winners (exact, sorted desc)
    if (wave == 0) {
        for (int k = 0; k < TOPK; ++k) {
            u64 best = 0;
            for (int j = lane; j < NWAVES * TOPK; j += 32) { u64 p = s_pk2[j]; best = p > best ? p : best; }
            #pragma unroll
            for (int off = 16; off > 0; off >>= 1) {
                u64 o = __shfl_xor(best, off, 32);
                best = o > best ? o : best;
            }
            if (lane == 0) {
                s_selv[k] = fkey_inv((unsigned)(best >> 32));
                s_selt[k] = NV - 1 - (int)(best & 0xFFFFFFFFull);
            }
            for (int j = lane; j < NWAVES * TOPK; j += 32)
                if (s_pk2[j] == best) s_pk2[j] = 0;
            __builtin_amdgcn_wave_barrier();
        }
    }
    __syncthreads();

    // temperature, softmax, inclusive cumsum, top-p mask (keep >= MINKEEP), re-softmax
    if (tid == 0) {
        const float temp = temperature[row];
        const float tp   = top_p[row];
        const float l0   = s_selv[0] / temp;      // sorted desc -> max logit
        float sum = 0.0f;
        for (int i = 0; i < TOPK; ++i) {
            float e = expf(s_selv[i] / temp - l0);
            s_e[i] = e;
            sum += e;
        }
        float cum = 0.0f, denom2 = 0.0f;
        for (int i = 0; i < TOPK; ++i) {
            cum += s_e[i] / sum;
            int keep = (cum < tp) || (i < MINKEEP);
            s_mk[i] = keep;
            if (keep) denom2 += s_e[i];
        }
        for (int i = 0; i < TOPK; ++i) {
            // masked-out: exp(-1000 - l0) underflows to exactly 0 in fp32
            out[row * TOPK + i]             = s_mk[i] ? (s_e[i] / denom2) : 0.0f;
            out[NB * TOPK + row * TOPK + i] = (float)s_selt[i];
        }
    }
}

extern "C" void kernel_launch(void* const* d_in, const int* in_sizes, int n_in,
                              void* d_out, int out_size, void* d_ws, size_t ws_size,
                              hipStream_t stream) {
    (void)in_sizes; (void)n_in; (void)out_size; (void)ws_size;
    const float* logits      = (const float*)d_in[0];
    const int*   ids         = (const int*)d_in[1];
    const float* top_p       = (const float*)d_in[2];
    const float* temperature = (const float*)d_in[3];
    const float* penalty     = (const float*)d_in[4];
    u64* pws = (u64*)d_ws;   // needs NSEGBLK*TOPK*8 = 1,024,000 bytes of scratch

    hipLaunchKernelGGL(seg_topk_kernel, dim3(NSEGBLK), dim3(NTHREADS), 0, stream,
                       logits, ids, penalty, pws);
    hipLaunchKernelGGL(merge_finalize_kernel, dim3(NB), dim3(NTHREADS), 0, stream,
                       pws, top_p, temperature, (float*)d_out);
}